// GATLayer_36928128811056
// MI455X (gfx1250) — compile-verified
//
#include <hip/hip_runtime.h>
#include <hip/hip_bf16.h>

typedef float v2f __attribute__((ext_vector_type(2)));
typedef float v8f __attribute__((ext_vector_type(8)));

#define IN_DIM 256
#define OUT_DIM 128
#define LEAKY_SLOPE 0.01f

// ---------------------------------------------------------------------------
// Kernel 0: zero-fill output (harness poisons d_out; segment-sum needs zeros)
// ---------------------------------------------------------------------------
__global__ void zero_f4(float4* __restrict__ p, int n4) {
    int i = blockIdx.x * blockDim.x + threadIdx.x;
    if (i < n4) p[i] = make_float4(0.f, 0.f, 0.f, 0.f);
}

// ---------------------------------------------------------------------------
// Kernel 1: z = h @ fc_w^T + fc_b via V_WMMA_F32_16X16X4_F32 (fp32 exact).
// Block = 256 threads = 8 waves; wave w computes the 16x16 tile at
// rows [16*blockIdx.x, +16), cols [16*w, +16).  K looped in steps of 4.
//
// A (16x4 f32, 2 VGPRs): lane L<16  -> row tm+L,    holds K=k,k+1
//                        lane L>=16 -> row tm+L-16, holds K=k+2,k+3
// B (4x16 f32, 2 VGPRs): lane L<16  -> col tn+L,    holds K=k,k+1  (B[k][n]=fc_w[n][k])
//                        lane L>=16 -> col tn+L-16, holds K=k+2,k+3
// C (16x16 f32, 8 VGPRs): VGPR r -> row tm + (L>=16 ? 8 : 0) + r, col tn + (L&15)
// ---------------------------------------------------------------------------
__global__ void gemm_wmma_f32(const float* __restrict__ h,
                              const float* __restrict__ fcw,
                              const float* __restrict__ fcb,
                              float* __restrict__ z,
                              int n_nodes) {
    const int wave = threadIdx.x >> 5;       // 0..7 -> N tile
    const int lane = threadIdx.x & 31;
    const int l16  = lane & 15;
    const int hi   = lane >> 4;              // 0 or 1 (K half / row half)

    const int tm = blockIdx.x * 16;
    const int tn = wave * 16;
    if (tm >= n_nodes) return;

    int arow = tm + l16;
    if (arow >= n_nodes) arow = n_nodes - 1;   // clamp for ragged tail

    const float* aptr = h   + (size_t)arow * IN_DIM       + hi * 2;
    const float* bptr = fcw + (size_t)(tn + l16) * IN_DIM + hi * 2;

    v8f acc = {};
    #pragma unroll 8
    for (int k = 0; k < IN_DIM; k += 4) {
        v2f a = *(const v2f*)(aptr + k);   // global_load_b64
        v2f b = *(const v2f*)(bptr + k);   // global_load_b64 (L2-resident, 128 KB)
        acc = __builtin_amdgcn_wmma_f32_16x16x4_f32(
            /*neg_a=*/false, a, /*neg_b=*/false, b,
            /*c_mod=*/(short)0, acc, /*reuse_a=*/false, /*reuse_b=*/false);
    }

    const float bias = fcb[tn + l16];
    const int row0 = tm + hi * 8;
    float* zp = z + (size_t)row0 * OUT_DIM + tn + l16;

    if (tm + 16 <= n_nodes) {
        // full tile: branch-free epilogue (the common case; N % 16 == 0 here)
        #pragma unroll
        for (int r = 0; r < 8; ++r) {
            zp[(size_t)r * OUT_DIM] = acc[r] + bias;
        }
    } else {
        #pragma unroll
        for (int r = 0; r < 8; ++r) {
            if (row0 + r < n_nodes) zp[(size_t)r * OUT_DIM] = acc[r] + bias;
        }
    }
}

// ---------------------------------------------------------------------------
// Kernel 2: per-node attention scalars
//   a_src[n] = dot(z[n], attn_w[0:128]) + attn_b   (bias folded in)
//   a_dst[n] = dot(z[n], attn_w[128:256])
// One wave per node; lane c handles channels 4c..4c+3; shuffle reduce.
// ---------------------------------------------------------------------------
__global__ void node_attn(const float* __restrict__ z,
                          const float* __restrict__ attn_w,
                          const float* __restrict__ attn_b,
                          float* __restrict__ a_src,
                          float* __restrict__ a_dst,
                          int n_nodes) {
    const int node = (blockIdx.x * blockDim.x + threadIdx.x) >> 5;
    const int lane = threadIdx.x & 31;
    if (node >= n_nodes) return;

    float4 zv = ((const float4*)(z + (size_t)node * OUT_DIM))[lane];
    float4 ws = ((const float4*)attn_w)[lane];
    float4 wd = ((const float4*)(attn_w + OUT_DIM))[lane];

    float s = zv.x * ws.x + zv.y * ws.y + zv.z * ws.z + zv.w * ws.w;
    float d = zv.x * wd.x + zv.y * wd.y + zv.z * wd.z + zv.w * wd.w;

    #pragma unroll
    for (int off = 16; off > 0; off >>= 1) {
        s += __shfl_down(s, off, 32);
        d += __shfl_down(d, off, 32);
    }
    if (lane == 0) {
        a_src[node] = s + attn_b[0];
        a_dst[node] = d;
    }
}

// ---------------------------------------------------------------------------
// Kernel 3: edge phase.  One wave per edge:
//   att  = a_src[s] + a_dst[d]          (bias already folded into a_src)
//   attr = leaky_relu(att)
//   out[d, :] += z[s, :] * attr         (128 f32 atomics per edge, 4 per lane)
// z/out/edge_index are all L2-resident (51.2+51.2+12.8 MB < 192 MB L2).
// ---------------------------------------------------------------------------
__global__ void edge_scatter(const int* __restrict__ ei,   // [2, E] flat
                             const float* __restrict__ z,
                             const float* __restrict__ a_src,
                             const float* __restrict__ a_dst,
                             float* __restrict__ out,
                             int num_edges) {
    const int e    = (blockIdx.x * blockDim.x + threadIdx.x) >> 5;
    const int lane = threadIdx.x & 31;
    if (e >= num_edges) return;

    const int s = ei[e];                 // same addr across wave -> 1 L2 request
    const int d = ei[num_edges + e];

    float att = a_src[s] + a_dst[d];
    float attr = att >= 0.f ? att : LEAKY_SLOPE * att;

    float4 zv = ((const float4*)(z + (size_t)s * OUT_DIM))[lane];  // global_load_b128
    float* o = out + (size_t)d * OUT_DIM + lane * 4;
    atomicAdd(o + 0, zv.x * attr);   // global_atomic_add_f32 (no return)
    atomicAdd(o + 1, zv.y * attr);
    atomicAdd(o + 2, zv.z * attr);
    atomicAdd(o + 3, zv.w * attr);
}

// ---------------------------------------------------------------------------
extern "C" void kernel_launch(void* const* d_in, const int* in_sizes, int n_in,
                              void* d_out, int out_size, void* d_ws, size_t ws_size,
                              hipStream_t stream) {
    const float* h      = (const float*)d_in[0];
    const int*   ei     = (const int*)  d_in[1];
    const float* fc_w   = (const float*)d_in[2];
    const float* fc_b   = (const float*)d_in[3];
    const float* attn_w = (const float*)d_in[4];
    const float* attn_b = (const float*)d_in[5];

    const int n_nodes   = in_sizes[0] / IN_DIM;   // 100000
    const int num_edges = in_sizes[1] / 2;        // 1600000

    // workspace layout
    float* z     = (float*)d_ws;                           // n_nodes * 128 f32
    float* a_src = z + (size_t)n_nodes * OUT_DIM;          // n_nodes f32
    float* a_dst = a_src + n_nodes;                        // n_nodes f32
    float* out   = (float*)d_out;

    // 0) out = 0
    {
        int n4 = (n_nodes * OUT_DIM) / 4;
        zero_f4<<<(n4 + 255) / 256, 256, 0, stream>>>((float4*)out, n4);
    }
    // 1) z = h @ fc_w^T + fc_b   (WMMA fp32)
    {
        int mblocks = (n_nodes + 15) / 16;       // 6250; 8 waves cover the 128 cols
        gemm_wmma_f32<<<mblocks, 256, 0, stream>>>(h, fc_w, fc_b, z, n_nodes);
    }
    // 2) per-node attention scalars
    {
        int blocks = (n_nodes + 7) / 8;          // 8 waves (nodes) per block
        node_attn<<<blocks, 256, 0, stream>>>(z, attn_w, attn_b, a_src, a_dst, n_nodes);
    }
    // 3) edge gather + leaky-relu scale + atomic scatter-add
    {
        int blocks = (num_edges + 7) / 8;        // 8 waves (edges) per block
        edge_scatter<<<blocks, 256, 0, stream>>>(ei, z, a_src, a_dst, out, num_edges);
    }
}